// GCN_62801011802129
// MI455X (gfx1250) — compile-verified
//
#include <hip/hip_runtime.h>

// GCN 2-layer: aggregate -> GEMM(relu) -> GEMM -> aggregate (linear-commuted) -> finalize
// FP32 WMMA (V_WMMA_F32_16X16X4_F32) for both GEMMs, fused in one kernel via LDS.

typedef __attribute__((ext_vector_type(2))) float v2f;
typedef __attribute__((ext_vector_type(8))) float v8f;

#define GCN_IN  128
#define GCN_HID 256
#define GCN_OUT 128

// ---------------- degree kernels ----------------
__global__ void k_init_deg(float* od, float* id, int n) {
    int i = blockIdx.x * blockDim.x + threadIdx.x;
    if (i < n) { od[i] = 1.0f; id[i] = 1.0f; }   // self-loop contributes 1 to both
}

__global__ void k_deg(const int* __restrict__ src, const int* __restrict__ dst,
                      float* od, float* id, int e) {
    int i = blockIdx.x * blockDim.x + threadIdx.x;
    if (i < e) {
        atomicAdd(&od[src[i]], 1.0f);
        atomicAdd(&id[dst[i]], 1.0f);
    }
}

__global__ void k_isqrt(float* od, float* id, int n) {
    int i = blockIdx.x * blockDim.x + threadIdx.x;
    if (i < n) {                       // deg >= 1 always, so max(deg,1)==deg
        od[i] = rsqrtf(od[i]);
        id[i] = rsqrtf(id[i]);
    }
}

// ---------------- layer-1 aggregation ----------------
// agg1[i] = x[i] * out_isqrt[i]   (self-loop term, no atomics needed)
__global__ void k_init_agg1(const float* __restrict__ x, const float* __restrict__ os,
                            float* __restrict__ agg, int n) {
    int gid = blockIdx.x * blockDim.x + threadIdx.x;   // n*32 threads, float4 each
    int node = gid >> 5, l = gid & 31;
    if (node < n) {
        float4 v = ((const float4*)x)[(size_t)node * 32 + l];
        float s = os[node];
        v.x *= s; v.y *= s; v.z *= s; v.w *= s;
        ((float4*)agg)[(size_t)node * 32 + l] = v;
    }
}

// one wave per edge: agg1[dst] += x[src] * out_isqrt[src]
__global__ void k_agg1(const float* __restrict__ x, const int* __restrict__ src,
                       const int* __restrict__ dst, const float* __restrict__ os,
                       float* __restrict__ agg, int e) {
    int gid = blockIdx.x * blockDim.x + threadIdx.x;
    int edge = gid >> 5, l = gid & 31;
    if (edge < e) {
        int s = src[edge], d = dst[edge];
        float sc = os[s];
        float4 v = ((const float4*)x)[(size_t)s * 32 + l];
        float* p = &agg[(size_t)d * GCN_IN + l * 4];
        atomicAdd(p + 0, v.x * sc);
        atomicAdd(p + 1, v.y * sc);
        atomicAdd(p + 2, v.z * sc);
        atomicAdd(p + 3, v.w * sc);
    }
}

// ---------------- fused MLP: both GEMMs with f32 WMMA ----------------
// One block = 16-row slab. 512 threads = 16 waves.
// Phase 1: wave w computes h1 tile cols [16w,16w+16), K=128, into LDS (relu * out_isqrt applied).
// Phase 2: waves 0..7 compute t tile cols [16w,16w+16), K=256 from LDS vs W2.
// N must be a multiple of 16 (50000 = 3125*16).
__global__ void __launch_bounds__(512)
k_fused_mlp(const float* __restrict__ agg1, const float* __restrict__ in_is,
            const float* __restrict__ out_is, const float* __restrict__ W1,
            const float* __restrict__ b1, const float* __restrict__ W2,
            float* __restrict__ t, float* __restrict__ out) {
    __shared__ float g_tile[16][260];            // padded: stride 260 -> conflict-free

    const int r0   = blockIdx.x * 16;
    const int wave = threadIdx.x >> 5;
    const int lane = threadIdx.x & 31;
    const int m    = lane & 15;                  // row within tile (A) / col within tile (B,D)
    const int kk   = (lane >> 4) << 1;           // K sub-offset: 0 (lanes 0-15) or 2 (16-31)
    const int drow = (lane >> 4) * 8;            // D: lanes 16-31 hold rows +8

    // ---- phase 1: h1 = (agg1 * in_isqrt) @ W1 + b1 ----
    {
        const int c0 = wave * 16;
        const float rs = in_is[r0 + m];
        const float* arow = agg1 + (size_t)(r0 + m) * GCN_IN;
        v8f acc = {};
        #pragma unroll 4
        for (int k = 0; k < GCN_IN; k += 4) {
            v2f a, b;
            a.x = arow[k + kk]     * rs;
            a.y = arow[k + kk + 1] * rs;
            b.x = W1[(size_t)(k + kk)     * GCN_HID + c0 + m];
            b.y = W1[(size_t)(k + kk + 1) * GCN_HID + c0 + m];
            acc = __builtin_amdgcn_wmma_f32_16x16x4_f32(
                false, a, false, b, (short)0, acc, false, false);
        }
        const float bias = b1[c0 + m];
        #pragma unroll
        for (int j = 0; j < 8; ++j) {
            int row = drow + j;
            float h = fmaxf(acc[j] + bias, 0.0f);         // relu
            g_tile[row][c0 + m] = h * out_is[r0 + row];   // fold layer-2 source scale
        }
    }
    __syncthreads();

    // ---- phase 2: t = g @ W2  (aggregation commuted past the linear layer) ----
    if (wave < 8) {                               // wave-uniform: EXEC stays all-ones
        const int c0 = wave * 16;
        v8f acc = {};
        #pragma unroll 4
        for (int k = 0; k < GCN_HID; k += 4) {
            v2f a, b;
            a.x = g_tile[m][k + kk];
            a.y = g_tile[m][k + kk + 1];
            b.x = W2[(size_t)(k + kk)     * GCN_OUT + c0 + m];
            b.y = W2[(size_t)(k + kk + 1) * GCN_OUT + c0 + m];
            acc = __builtin_amdgcn_wmma_f32_16x16x4_f32(
                false, a, false, b, (short)0, acc, false, false);
        }
        #pragma unroll
        for (int j = 0; j < 8; ++j) {
            int row = r0 + drow + j;
            float v = acc[j];
            t  [(size_t)row * GCN_OUT + c0 + m] = v;
            out[(size_t)row * GCN_OUT + c0 + m] = v;      // self-loop init of layer-2 agg
        }
    }
}

// ---------------- layer-2 aggregation over t (128-wide) ----------------
__global__ void k_agg2(const float* __restrict__ t, const int* __restrict__ src,
                       const int* __restrict__ dst, float* __restrict__ out, int e) {
    int gid = blockIdx.x * blockDim.x + threadIdx.x;
    int edge = gid >> 5, l = gid & 31;
    if (edge < e) {
        int s = src[edge], d = dst[edge];
        float4 v = ((const float4*)t)[(size_t)s * 32 + l];
        float* p = &out[(size_t)d * GCN_OUT + l * 4];
        atomicAdd(p + 0, v.x);
        atomicAdd(p + 1, v.y);
        atomicAdd(p + 2, v.z);
        atomicAdd(p + 3, v.w);
    }
}

// out = out * in_isqrt[row] + b2
__global__ void k_final(float* __restrict__ out, const float* __restrict__ is,
                        const float* __restrict__ b2, int n) {
    int gid = blockIdx.x * blockDim.x + threadIdx.x;
    int node = gid >> 5, l = gid & 31;
    if (node < n) {
        float4 v = ((float4*)out)[(size_t)node * 32 + l];
        float4 bb = ((const float4*)b2)[l];
        float s = is[node];
        v.x = v.x * s + bb.x;
        v.y = v.y * s + bb.y;
        v.z = v.z * s + bb.z;
        v.w = v.w * s + bb.w;
        ((float4*)out)[(size_t)node * 32 + l] = v;
    }
}

extern "C" void kernel_launch(void* const* d_in, const int* in_sizes, int n_in,
                              void* d_out, int out_size, void* d_ws, size_t ws_size,
                              hipStream_t stream) {
    const float* x  = (const float*)d_in[0];
    const int*   ei = (const int*)d_in[1];
    const float* W1 = (const float*)d_in[2];
    const float* b1 = (const float*)d_in[3];
    const float* W2 = (const float*)d_in[4];
    const float* b2 = (const float*)d_in[5];

    const int N = in_sizes[0] / GCN_IN;     // 50000
    const int E = in_sizes[1] / 2;          // 800000
    const int* src = ei;
    const int* dst = ei + E;

    // workspace layout (floats): out_isqrt[N] | in_isqrt[N] | agg1[N*128] | t[N*128]
    float* od = (float*)d_ws;
    float* idg = od + N;
    size_t off = ((size_t)(2 * N) + 255) & ~(size_t)255;
    float* agg1 = (float*)d_ws + off;
    float* t    = agg1 + (size_t)N * GCN_IN;
    float* out  = (float*)d_out;

    const int B = 256;
    const long nTot = (long)N * 32;         // node-feature float4 threads
    const long eTot = (long)E * 32;         // edge-feature float4 threads

    k_init_deg <<<(N + B - 1) / B, B, 0, stream>>>(od, idg, N);
    k_deg      <<<(E + B - 1) / B, B, 0, stream>>>(src, dst, od, idg, E);
    k_isqrt    <<<(N + B - 1) / B, B, 0, stream>>>(od, idg, N);
    k_init_agg1<<<(nTot + B - 1) / B, B, 0, stream>>>(x, od, agg1, N);
    k_agg1     <<<(eTot + B - 1) / B, B, 0, stream>>>(x, src, dst, od, agg1, E);
    k_fused_mlp<<<N / 16, 512, 0, stream>>>(agg1, idg, od, W1, b1, W2, t, out);
    k_agg2     <<<(eTot + B - 1) / B, B, 0, stream>>>(t, src, dst, out, E);
    k_final    <<<(nTot + B - 1) / B, B, 0, stream>>>(out, idg, b2, N);
}